// SetAbstraction_81913616270004
// MI455X (gfx1250) — compile-verified
//
#include <hip/hip_runtime.h>
#include <hip/hip_bf16.h>
#include <stdint.h>

// Problem constants (match reference)
#define B_   4
#define N_   16384
#define S_   2048
#define K_   32
#define NC   (B_ * S_ * K_)      // 262144 GEMM columns, col = (b*S + s)*K + k
#define RADIUS2 0.04f            // 0.2^2
#define EPS_ 1e-5f
#define XYZ_OUT (B_ * 3 * S_)    // 24576 floats: first output (new_xyz transposed)

typedef __attribute__((ext_vector_type(2))) float v2f;
typedef __attribute__((ext_vector_type(8))) float v8f;

// ---------------------------------------------------------------------------
// Kernel 1: farthest point sampling. One block per batch, dist[] in registers
// (16 values/thread @ 1024 threads), LDS argmax tree. Emits new_xyz both in
// (B,S,3) scratch layout (for ball query) and (B,3,S) into d_out.
// ---------------------------------------------------------------------------
__global__ __launch_bounds__(1024) void fps_kernel(const float* __restrict__ pos,
                                                   float* __restrict__ nxyz,
                                                   float* __restrict__ out_xyz) {
  const int b = blockIdx.x;
  const int t = threadIdx.x;
  const float* p = pos + (size_t)b * N_ * 3;

  __shared__ float rv[1024];
  __shared__ int   ri[1024];
  __shared__ int   sfar;

  float dist[16];
#pragma unroll
  for (int j = 0; j < 16; ++j) dist[j] = 1e10f;

  int far = 0;
  for (int s = 0; s < S_; ++s) {
    const float cx = p[far * 3 + 0];
    const float cy = p[far * 3 + 1];
    const float cz = p[far * 3 + 2];
    if (t == 0) {
      const size_t ws = (size_t)b * S_ + s;
      nxyz[ws * 3 + 0] = cx; nxyz[ws * 3 + 1] = cy; nxyz[ws * 3 + 2] = cz;
      out_xyz[((size_t)b * 3 + 0) * S_ + s] = cx;
      out_xyz[((size_t)b * 3 + 1) * S_ + s] = cy;
      out_xyz[((size_t)b * 3 + 2) * S_ + s] = cz;
    }
    float bestv = -1.0f;
    int   besti = 0;
#pragma unroll
    for (int j = 0; j < 16; ++j) {
      const int i = j * 1024 + t;              // coalesced, ascending index
      const float dx = p[i * 3 + 0] - cx;
      const float dy = p[i * 3 + 1] - cy;
      const float dz = p[i * 3 + 2] - cz;
      const float d = dx * dx + dy * dy + dz * dz;
      const float nd = fminf(dist[j], d);
      dist[j] = nd;
      if (nd > bestv) { bestv = nd; besti = i; } // strict > keeps lowest index
    }
    rv[t] = bestv; ri[t] = besti;
    __syncthreads();
    for (int off = 512; off > 0; off >>= 1) {
      if (t < off) {
        const float v2 = rv[t + off]; const int i2 = ri[t + off];
        if (v2 > rv[t] || (v2 == rv[t] && i2 < ri[t])) { rv[t] = v2; ri[t] = i2; }
      }
      __syncthreads();
    }
    if (t == 0) sfar = ri[0];
    __syncthreads();
    far = sfar;
  }
}

// ---------------------------------------------------------------------------
// Kernel 2: ball query + grouping. One wave32 per centroid; ballot/popc prefix
// selects the first K in-radius points in index order (early exit), writes the
// grouped tensor as x0[8][NC] (rows 6,7 zero-padded for the WMMA K=4 step).
// ---------------------------------------------------------------------------
__global__ __launch_bounds__(256) void group_kernel(const float* __restrict__ pos,
                                                    const float* __restrict__ feats,
                                                    const float* __restrict__ nxyz,
                                                    float* __restrict__ x0) {
  const int lane = threadIdx.x & 31;
  const int w = blockIdx.x * 8 + (threadIdx.x >> 5);   // centroid id in [0, B*S)
  const int b = w / S_;
  const float* p = pos   + (size_t)b * N_ * 3;
  const float* f = feats + (size_t)b * N_ * 3;
  const float cx = nxyz[(size_t)w * 3 + 0];
  const float cy = nxyz[(size_t)w * 3 + 1];
  const float cz = nxyz[(size_t)w * 3 + 2];
  const size_t colbase = (size_t)w * K_;

  int cnt = 0;
  int firstIdx = -1;
  for (int base = 0; base < N_ && cnt < K_; base += 32) {
    const int i = base + lane;
    const float dx = p[i * 3 + 0] - cx;
    const float dy = p[i * 3 + 1] - cy;
    const float dz = p[i * 3 + 2] - cz;
    const float d = dx * dx + dy * dy + dz * dz;
    const bool in = (d <= RADIUS2);
    const unsigned mask = (unsigned)__ballot(in);
    if (firstIdx < 0 && mask) firstIdx = base + __ffs(mask) - 1;
    const int posn = __popc(mask & ((1u << lane) - 1u));
    if (in && (cnt + posn) < K_) {
      const size_t col = colbase + cnt + posn;
      x0[0 * (size_t)NC + col] = dx;
      x0[1 * (size_t)NC + col] = dy;
      x0[2 * (size_t)NC + col] = dz;
      x0[3 * (size_t)NC + col] = f[i * 3 + 0];
      x0[4 * (size_t)NC + col] = f[i * 3 + 1];
      x0[5 * (size_t)NC + col] = f[i * 3 + 2];
      x0[6 * (size_t)NC + col] = 0.0f;
      x0[7 * (size_t)NC + col] = 0.0f;
    }
    cnt += __popc(mask);
  }
  if (cnt < K_) {                 // pad remaining slots with first neighbor
    const int i = firstIdx;
    const float dx = p[i * 3 + 0] - cx;
    const float dy = p[i * 3 + 1] - cy;
    const float dz = p[i * 3 + 2] - cz;
    const float f0 = f[i * 3 + 0], f1 = f[i * 3 + 1], f2 = f[i * 3 + 2];
    for (int k = cnt + lane; k < K_; k += 32) {
      const size_t col = colbase + k;
      x0[0 * (size_t)NC + col] = dx;
      x0[1 * (size_t)NC + col] = dy;
      x0[2 * (size_t)NC + col] = dz;
      x0[3 * (size_t)NC + col] = f0;
      x0[4 * (size_t)NC + col] = f1;
      x0[5 * (size_t)NC + col] = f2;
      x0[6 * (size_t)NC + col] = 0.0f;
      x0[7 * (size_t)NC + col] = 0.0f;
    }
  }
}

// ---------------------------------------------------------------------------
// Kernel 3: WMMA f32 GEMM  Y[O][NC] = W[O][C] * X[C][NC] + bias, with the
// previous layer's BN+ReLU fused into the X load. One wave computes a 16x16
// tile via V_WMMA_F32_16X16X4_F32; 8 waves/block tile 64(O) x 32(cols).
// ---------------------------------------------------------------------------
__global__ __launch_bounds__(256) void gemm_kernel(const float* __restrict__ W,
                                                   const float* __restrict__ bias,
                                                   const float* __restrict__ X,
                                                   const float* __restrict__ pScale,
                                                   const float* __restrict__ pShift,
                                                   int C, int applyPrev,
                                                   float* __restrict__ Y) {
  __shared__ float xs[128 * 33];               // C<=128 rows, 32 cols, +1 pad
  const int tid = threadIdx.x;
  const size_t colBase = (size_t)blockIdx.x * 32;
  const int oBase64 = blockIdx.y * 64;

  for (int idx = tid; idx < C * 32; idx += 256) {
    const int c = idx >> 5, n = idx & 31;
    float v = X[(size_t)c * NC + colBase + n];
    if (applyPrev) { v = fmaxf(fmaf(v, pScale[c], pShift[c]), 0.0f); }
    xs[c * 33 + n] = v;
  }
  __syncthreads();

  const int wave = tid >> 5, lane = tid & 31;
  const int oT = oBase64 + (wave >> 1) * 16;   // 16-row output tile
  const int nT = (wave & 1) * 16;              // 16-col output tile
  const int m  = lane & 15;
  const int hi = lane >> 4;

  v8f acc;
#pragma unroll
  for (int j = 0; j < 8; ++j) acc[j] = bias[oT + j + 8 * hi];

  const float* wrow = W + (size_t)(oT + m) * C;
  for (int c0 = 0; c0 < C; c0 += 4) {
    v2f a, bf;
    a.x  = wrow[c0 + 2 * hi];                  // A: 16x4, K = 2*hi, 2*hi+1
    a.y  = wrow[c0 + 2 * hi + 1];
    bf.x = xs[(c0 + 2 * hi) * 33 + nT + m];    // B: 4x16, same K split
    bf.y = xs[(c0 + 2 * hi + 1) * 33 + nT + m];
    acc = __builtin_amdgcn_wmma_f32_16x16x4_f32(false, a, false, bf,
                                                (short)0, acc, false, false);
  }
#pragma unroll
  for (int j = 0; j < 8; ++j)
    Y[(size_t)(oT + j + 8 * hi) * NC + colBase + nT + m] = acc[j];
}

// ---------------------------------------------------------------------------
// Kernel 4: per-channel partial sums (deterministic, no float atomics).
// grid = (32 chunks, O channels); fixed-order LDS tree per block.
// ---------------------------------------------------------------------------
__global__ __launch_bounds__(256) void stats_kernel(const float* __restrict__ Y,
                                                    float* __restrict__ psum,
                                                    float* __restrict__ psq) {
  const int o = blockIdx.y, chunk = blockIdx.x;
  const size_t base = (size_t)o * NC + (size_t)chunk * (NC / 32);
  float s = 0.0f, q = 0.0f;
  for (int i = threadIdx.x; i < NC / 32; i += 256) {
    const float v = Y[base + i];
    s += v; q += v * v;
  }
  __shared__ float ss[256], qs[256];
  ss[threadIdx.x] = s; qs[threadIdx.x] = q;
  __syncthreads();
  for (int off = 128; off > 0; off >>= 1) {
    if (threadIdx.x < off) {
      ss[threadIdx.x] += ss[threadIdx.x + off];
      qs[threadIdx.x] += qs[threadIdx.x + off];
    }
    __syncthreads();
  }
  if (threadIdx.x == 0) { psum[o * 32 + chunk] = ss[0]; psq[o * 32 + chunk] = qs[0]; }
}

// Fold partials (fixed order) into fused BN params: y = relu(x*scale + shift).
__global__ void params_kernel(const float* __restrict__ psum,
                              const float* __restrict__ psq,
                              const float* __restrict__ g,
                              const float* __restrict__ beta,
                              int O, float* __restrict__ scale,
                              float* __restrict__ shift) {
  const int o = blockIdx.x * blockDim.x + threadIdx.x;
  if (o >= O) return;
  float s = 0.0f, q = 0.0f;
  for (int i = 0; i < 32; ++i) { s += psum[o * 32 + i]; q += psq[o * 32 + i]; }
  const float mu  = s * (1.0f / NC);
  const float var = q * (1.0f / NC) - mu * mu;
  const float inv = rsqrtf(var + EPS_);
  const float sc  = g[o] * inv;
  scale[o] = sc;
  shift[o] = fmaf(-mu, sc, beta[o]);
}

// Zero-pad W1 (64x6) to (64x8) so layer 1 shares the WMMA K=4 pipeline.
__global__ void padw1_kernel(const float* __restrict__ w1, float* __restrict__ wpad) {
  const int i = blockIdx.x * blockDim.x + threadIdx.x;
  if (i < 64 * 8) {
    const int o = i >> 3, c = i & 7;
    wpad[i] = (c < 6) ? w1[o * 6 + c] : 0.0f;
  }
}

// ---------------------------------------------------------------------------
// Kernel 5: fused layer-3 BN+ReLU + max over K. Sign-aware (scale may be <0):
// track min & max of raw y3 over K, pick branch, then ReLU.
// ---------------------------------------------------------------------------
__global__ __launch_bounds__(256) void maxpool_kernel(const float* __restrict__ Y3,
                                                      const float* __restrict__ scale,
                                                      const float* __restrict__ shift,
                                                      float* __restrict__ out) {
  const size_t idx = (size_t)blockIdx.x * 256 + threadIdx.x;   // (b*256+o)*S+s
  const int s = (int)(idx % S_);
  const int o = (int)((idx / S_) % 256);
  const int b = (int)(idx / ((size_t)S_ * 256));
  const size_t base = (size_t)o * NC + ((size_t)b * S_ + s) * K_;
  const float4* p4 = (const float4*)(Y3 + base);
  float mx = -3.4e38f, mn = 3.4e38f;
#pragma unroll
  for (int i = 0; i < 8; ++i) {
    const float4 v = p4[i];
    mx = fmaxf(mx, fmaxf(fmaxf(v.x, v.y), fmaxf(v.z, v.w)));
    mn = fminf(mn, fminf(fminf(v.x, v.y), fminf(v.z, v.w)));
  }
  const float sc = scale[o], sh = shift[o];
  const float v = (sc >= 0.0f) ? fmaf(mx, sc, sh) : fmaf(mn, sc, sh);
  out[XYZ_OUT + idx] = fmaxf(v, 0.0f);
}

// ---------------------------------------------------------------------------
extern "C" void kernel_launch(void* const* d_in, const int* in_sizes, int n_in,
                              void* d_out, int out_size, void* d_ws, size_t ws_size,
                              hipStream_t stream) {
  const float* pos   = (const float*)d_in[0];
  const float* feats = (const float*)d_in[1];
  const float* w1 = (const float*)d_in[2];
  const float* b1 = (const float*)d_in[3];
  const float* g1 = (const float*)d_in[4];
  const float* be1 = (const float*)d_in[5];
  const float* w2 = (const float*)d_in[6];
  const float* b2 = (const float*)d_in[7];
  const float* g2 = (const float*)d_in[8];
  const float* be2 = (const float*)d_in[9];
  const float* w3 = (const float*)d_in[10];
  const float* b3 = (const float*)d_in[11];
  const float* g3 = (const float*)d_in[12];
  const float* be3 = (const float*)d_in[13];
  float* out = (float*)d_out;

  float* wsf = (float*)d_ws;
  size_t off = 0;
  float* x0   = wsf + off; off += (size_t)8   * NC;
  float* y1   = wsf + off; off += (size_t)64  * NC;
  float* y2   = wsf + off; off += (size_t)128 * NC;
  float* y3   = wsf + off; off += (size_t)256 * NC;
  float* nxyz = wsf + off; off += (size_t)B_ * S_ * 3;
  float* wpad = wsf + off; off += 64 * 8;
  float* psum = wsf + off; off += 256 * 32;
  float* psq  = wsf + off; off += 256 * 32;
  float* sc1  = wsf + off; off += 64;
  float* sh1  = wsf + off; off += 64;
  float* sc2  = wsf + off; off += 128;
  float* sh2  = wsf + off; off += 128;
  float* sc3  = wsf + off; off += 256;
  float* sh3  = wsf + off; off += 256;

  padw1_kernel<<<1, 512, 0, stream>>>(w1, wpad);
  fps_kernel<<<B_, 1024, 0, stream>>>(pos, nxyz, out);
  group_kernel<<<(B_ * S_) / 8, 256, 0, stream>>>(pos, feats, nxyz, x0);

  // Layer 1: 64 <- 8(padded 6)
  gemm_kernel<<<dim3(NC / 32, 1), 256, 0, stream>>>(wpad, b1, x0, nullptr, nullptr, 8, 0, y1);
  stats_kernel<<<dim3(32, 64), 256, 0, stream>>>(y1, psum, psq);
  params_kernel<<<1, 64, 0, stream>>>(psum, psq, g1, be1, 64, sc1, sh1);

  // Layer 2: 128 <- 64 (BN1+ReLU fused into load)
  gemm_kernel<<<dim3(NC / 32, 2), 256, 0, stream>>>(w2, b2, y1, sc1, sh1, 64, 1, y2);
  stats_kernel<<<dim3(32, 128), 256, 0, stream>>>(y2, psum, psq);
  params_kernel<<<2, 64, 0, stream>>>(psum, psq, g2, be2, 128, sc2, sh2);

  // Layer 3: 256 <- 128 (BN2+ReLU fused into load)
  gemm_kernel<<<dim3(NC / 32, 4), 256, 0, stream>>>(w3, b3, y2, sc2, sh2, 128, 1, y3);
  stats_kernel<<<dim3(32, 256), 256, 0, stream>>>(y3, psum, psq);
  params_kernel<<<4, 64, 0, stream>>>(psum, psq, g3, be3, 256, sc3, sh3);

  // BN3+ReLU fused into max over K
  maxpool_kernel<<<(B_ * 256 * S_) / 256, 256, 0, stream>>>(y3, sc3, sh3, out);
}